// SURVEY_shapes_GCN_81638738363110
// MI455X (gfx1250) — compile-verified
//
#include <hip/hip_runtime.h>

typedef __attribute__((ext_vector_type(2))) float v2f;
typedef __attribute__((ext_vector_type(8))) float v8f;

#define LDF 32  // padded feature stride (128 B)

__global__ void k_init_deg(float* __restrict__ deg, int n) {
  int i = blockIdx.x * blockDim.x + threadIdx.x;
  if (i < n) deg[i] = 1.0f;
}

template <typename IdxT>
__global__ void k_accum_deg(float* __restrict__ deg, const IdxT* __restrict__ dst,
                            const float* __restrict__ w, int E) {
  int e = blockIdx.x * blockDim.x + threadIdx.x;
  if (e < E) atomicAdd(&deg[(int)dst[e]], w[e]);
}

__global__ void k_finalize_dinv(float* __restrict__ deg, int n) {
  int i = blockIdx.x * blockDim.x + threadIdx.x;
  if (i < n) {
    float v = deg[i];
    deg[i] = v > 0.0f ? rsqrtf(v) : 0.0f;
  }
}

template <typename IdxT>
__global__ void k_norm(float* __restrict__ norm, const float* __restrict__ dinv,
                       const IdxT* __restrict__ src, const IdxT* __restrict__ dst,
                       const float* __restrict__ w, int E) {
  int e = blockIdx.x * blockDim.x + threadIdx.x;
  if (e < E) norm[e] = dinv[(int)src[e]] * w[e] * dinv[(int)dst[e]];
}

__global__ void k_cvt_idx(const long long* __restrict__ ei, int* __restrict__ s32,
                          int* __restrict__ d32, int E) {
  int e = blockIdx.x * blockDim.x + threadIdx.x;
  if (e < E) {
    s32[e] = (int)ei[e];
    d32[e] = (int)ei[(long long)E + e];
  }
}

__global__ void k_pad_x(const float* __restrict__ x, float* __restrict__ xp,
                        int n, int F) {
  int idx = blockIdx.x * blockDim.x + threadIdx.x;
  if (idx < n * LDF) {
    int i = idx >> 5, f = idx & (LDF - 1);
    xp[idx] = (f < F) ? x[i * F + f] : 0.0f;
  }
}

__global__ void k_pad_wb(const float* __restrict__ W, const float* __restrict__ b,
                         float* __restrict__ Wp, float* __restrict__ bp,
                         int K, int Nout, int Kp, int ldw) {
  int idx = blockIdx.x * blockDim.x + threadIdx.x;
  if (idx < Kp * ldw) {
    int k = idx / ldw, c = idx - k * ldw;
    Wp[idx] = (k < K && c < Nout) ? W[k * Nout + c] : 0.0f;
  }
  if (idx < ldw && bp != nullptr) bp[idx] = (idx < Nout) ? b[idx] : 0.0f;
}

__global__ void k_zero(float* __restrict__ p, int n) {
  int i = blockIdx.x * blockDim.x + threadIdx.x;
  if (i < n) p[i] = 0.0f;
}

__global__ void k_init_out(const float4* __restrict__ h4, const float* __restrict__ dinv,
                           const float4* __restrict__ biasP4, float4* __restrict__ out4,
                           int n) {
  int idx = blockIdx.x * blockDim.x + threadIdx.x;
  if (idx < n * (LDF / 4)) {
    int i = idx >> 3, q = idx & 7;
    float di = dinv[i];
    float s = di * di;
    float4 hv = h4[idx], bv = biasP4[q];
    float4 r;
    r.x = bv.x + hv.x * s;
    r.y = bv.y + hv.y * s;
    r.z = bv.z + hv.z * s;
    r.w = bv.w + hv.w * s;
    out4[idx] = r;
  }
}

template <typename IdxT>
__global__ void k_scatter(const float* __restrict__ h, const float* __restrict__ norm,
                          const IdxT* __restrict__ src, const IdxT* __restrict__ dst,
                          int E, float* __restrict__ out) {
  int e = blockIdx.x * blockDim.x + threadIdx.x;
  if (e >= E) return;
  int s = (int)src[e];
  int d = (int)dst[e];
  float nr = norm[e];
  const float4* __restrict__ hs = (const float4*)(h + (size_t)s * LDF);
  float* __restrict__ od = out + (size_t)d * LDF;
#pragma unroll
  for (int q = 0; q < 7; ++q) {
    float4 v = hs[q];
    atomicAdd(&od[q * 4 + 0], v.x * nr);
    atomicAdd(&od[q * 4 + 1], v.y * nr);
    atomicAdd(&od[q * 4 + 2], v.z * nr);
    atomicAdd(&od[q * 4 + 3], v.w * nr);
  }
  float4 v = hs[7];  // cols 28..31 (30,31 are zero pad)
  atomicAdd(&od[28], v.x * nr);
  atomicAdd(&od[29], v.y * nr);
}

__global__ void k_relu4(float4* __restrict__ x, int n4) {
  int i = blockIdx.x * blockDim.x + threadIdx.x;
  if (i < n4) {
    float4 v = x[i];
    v.x = fmaxf(v.x, 0.0f);
    v.y = fmaxf(v.y, 0.0f);
    v.z = fmaxf(v.z, 0.0f);
    v.w = fmaxf(v.w, 0.0f);
    x[i] = v;
  }
}

// ---------------------------------------------------------------------------
// WMMA fp32 GEMM: branchless K-unrolled core (V_WMMA_F32_16X16X4_F32),
// wave-uniform full-store epilogue, pure 32-bit addressing.
// ---------------------------------------------------------------------------
template <int KP>
__global__ void k_gemm_wmma(const float* __restrict__ X, int lda,
                            const float* __restrict__ Wp, int ldw,
                            const float* __restrict__ biasP,
                            float* __restrict__ Y, int ldy, int noutStore,
                            int M, int n_tiles, int do_relu) {
  int wid = blockIdx.x * (blockDim.x >> 5) + (threadIdx.x >> 5);
  int m_tiles = (M + 15) >> 4;
  if (wid >= m_tiles * n_tiles) return;  // wave-uniform exit
  int mt = wid / n_tiles;
  int nt = wid - mt * n_tiles;

  int lane = threadIdx.x & 31;
  int half = lane >> 4;
  int l16  = lane & 15;
  int row  = mt * 16 + l16;
  int rowc = row < M ? row : M - 1;  // branchless clamp (load-safe)
  int col  = nt * 16 + l16;

  const float* __restrict__ Xr = X + rowc * lda + half * 2;
  const float* __restrict__ Wc = Wp + (half * 2) * ldw + col;

  v8f acc = {};
#pragma unroll
  for (int k0 = 0; k0 < KP; k0 += 4) {
    v2f a = *(const v2f*)(Xr + k0);
    v2f b;
    b.x = Wc[k0 * ldw];
    b.y = Wc[(k0 + 1) * ldw];
    acc = __builtin_amdgcn_wmma_f32_16x16x4_f32(false, a, false, b,
                                                (short)0, acc, false, false);
  }

  // fold bias (+relu) into acc with wave-uniform branch (no per-elem cndmask)
  float bv = biasP[col];
  if (do_relu) {
#pragma unroll
    for (int v = 0; v < 8; ++v) acc[v] = fmaxf(acc[v] + bv, 0.0f);
  } else {
#pragma unroll
    for (int v = 0; v < 8; ++v) acc[v] = acc[v] + bv;
  }

  // C/D layout: VGPR v -> row v (lanes 0-15) / row v+8 (lanes 16-31)
  float* __restrict__ Yp = Y + (mt * 16 + half * 8) * ldy + col;
  bool fullStore = (mt * 16 + 16 <= M) & ((nt + 1) * 16 <= noutStore);
  if (fullStore) {  // wave-uniform: unconditional stores, no EXEC churn
#pragma unroll
    for (int v = 0; v < 8; ++v) {
      Yp[0] = acc[v];
      Yp += ldy;
    }
  } else {
    bool colOk = col < noutStore;
#pragma unroll
    for (int v = 0; v < 8; ++v) {
      int m = mt * 16 + half * 8 + v;
      if (m < M && colOk) Yp[0] = acc[v];
      Yp += ldy;
    }
  }
}

template <int KP>
static inline void launch_gemm(const float* X, int lda, const float* Wp, int ldw,
                               const float* biasP, float* Y, int ldy, int noutStore,
                               int M, int relu, hipStream_t s) {
  int m_tiles = (M + 15) / 16;
  int n_tiles = ldw / 16;
  long long waves = (long long)m_tiles * n_tiles;
  int blocks = (int)((waves + 7) / 8);  // 256 threads = 8 wave32 per block
  k_gemm_wmma<KP><<<blocks, 256, 0, s>>>(X, lda, Wp, ldw, biasP, Y, ldy,
                                         noutStore, M, n_tiles, relu);
}

extern "C" void kernel_launch(void* const* d_in, const int* in_sizes, int n_in,
                              void* d_out, int out_size, void* d_ws, size_t ws_size,
                              hipStream_t stream) {
  const float*     x   = (const float*)d_in[0];
  const long long* ei  = (const long long*)d_in[1];
  const float*     w   = (const float*)d_in[2];
  const float*     W1  = (const float*)d_in[3];
  const float*     b1  = (const float*)d_in[4];
  const float*     W2  = (const float*)d_in[5];
  const float*     b2  = (const float*)d_in[6];
  const float*     W3  = (const float*)d_in[7];
  const float*     b3  = (const float*)d_in[8];
  const float*     lw1 = (const float*)d_in[9];
  const float*     lb1 = (const float*)d_in[10];
  const float*     lw2 = (const float*)d_in[11];
  const float*     lb2 = (const float*)d_in[12];
  float* out = (float*)d_out;

  const int N = in_sizes[0] / 10;
  const int E = in_sizes[2];
  const int T = 256;

  char* ws = (char*)d_ws;
  size_t off = 0;
  auto alloc = [&](size_t bytes) -> char* {
    char* p = ws + off;
    off += (bytes + 255) & ~(size_t)255;
    return p;
  };
  float* dinv = (float*)alloc((size_t)N * 4);
  float* norm = (float*)alloc((size_t)E * 4);
  float* bufX = (float*)alloc((size_t)N * LDF * 4);
  float* bufA = (float*)alloc((size_t)N * LDF * 4);
  float* bufB = (float*)alloc((size_t)N * LDF * 4);
  float* Wp1 = (float*)alloc(12 * 32 * 4);
  float* Wp2 = (float*)alloc(32 * 32 * 4);
  float* Wp3 = (float*)alloc(32 * 32 * 4);
  float* Wp4 = (float*)alloc(32 * 16 * 4);
  float* Wp5 = (float*)alloc(12 * 16 * 4);
  float* b1p  = (float*)alloc(32 * 4);
  float* b2p  = (float*)alloc(32 * 4);
  float* b3p  = (float*)alloc(32 * 4);
  float* lb1p = (float*)alloc(16 * 4);
  float* lb2p = (float*)alloc(16 * 4);
  float* zb   = (float*)alloc(32 * 4);  // zero bias for conv GEMMs
  int* s32 = (int*)alloc((size_t)E * 4);
  int* d32 = (int*)alloc((size_t)E * 4);
  const bool use32 = (ws_size >= off);

  int gE  = (E + T - 1) / T;
  int gN  = (N + T - 1) / T;
  int gNF = (N * LDF + T - 1) / T;
  int gN4 = (N * (LDF / 4) + T - 1) / T;

  // one-time padded copies
  k_pad_x<<<gNF, T, 0, stream>>>(x, bufX, N, 10);
  k_pad_wb<<<2, T, 0, stream>>>(W1, b1, Wp1, b1p, 10, 30, 12, 32);
  k_pad_wb<<<4, T, 0, stream>>>(W2, b2, Wp2, b2p, 30, 30, 32, 32);
  k_pad_wb<<<4, T, 0, stream>>>(W3, b3, Wp3, b3p, 30, 30, 32, 32);
  k_pad_wb<<<2, T, 0, stream>>>(lw1, lb1, Wp4, lb1p, 30, 10, 32, 16);
  k_pad_wb<<<1, T, 0, stream>>>(lw2, lb2, Wp5, lb2p, 10, 4, 12, 16);
  k_zero<<<1, 32, 0, stream>>>(zb, 32);

  // normalization: deg (incl. self-loop), dinv, per-edge norm
  k_init_deg<<<gN, T, 0, stream>>>(dinv, N);
  if (use32) {
    k_cvt_idx<<<gE, T, 0, stream>>>(ei, s32, d32, E);
    k_accum_deg<int><<<gE, T, 0, stream>>>(dinv, d32, w, E);
    k_finalize_dinv<<<gN, T, 0, stream>>>(dinv, N);
    k_norm<int><<<gE, T, 0, stream>>>(norm, dinv, s32, d32, w, E);
  } else {
    k_accum_deg<long long><<<gE, T, 0, stream>>>(dinv, ei + E, w, E);
    k_finalize_dinv<<<gN, T, 0, stream>>>(dinv, N);
    k_norm<long long><<<gE, T, 0, stream>>>(norm, dinv, ei, ei + E, w, E);
  }

  // conv1: 10 -> 30 (no relu); bias + self-loop handled in k_init_out
  launch_gemm<12>(bufX, LDF, Wp1, 32, zb, bufA, LDF, LDF, N, 0, stream);
  k_init_out<<<gN4, T, 0, stream>>>((const float4*)bufA, dinv, (const float4*)b1p,
                                    (float4*)bufB, N);
  if (use32) k_scatter<int><<<gE, T, 0, stream>>>(bufA, norm, s32, d32, E, bufB);
  else       k_scatter<long long><<<gE, T, 0, stream>>>(bufA, norm, ei, ei + E, E, bufB);

  // conv2: 30 -> 30 (relu)
  launch_gemm<32>(bufB, LDF, Wp2, 32, zb, bufA, LDF, LDF, N, 0, stream);
  k_init_out<<<gN4, T, 0, stream>>>((const float4*)bufA, dinv, (const float4*)b2p,
                                    (float4*)bufB, N);
  if (use32) k_scatter<int><<<gE, T, 0, stream>>>(bufA, norm, s32, d32, E, bufB);
  else       k_scatter<long long><<<gE, T, 0, stream>>>(bufA, norm, ei, ei + E, E, bufB);
  k_relu4<<<gN4, T, 0, stream>>>((float4*)bufB, N * (LDF / 4));

  // conv3: 30 -> 30 (relu)
  launch_gemm<32>(bufB, LDF, Wp3, 32, zb, bufA, LDF, LDF, N, 0, stream);
  k_init_out<<<gN4, T, 0, stream>>>((const float4*)bufA, dinv, (const float4*)b3p,
                                    (float4*)bufB, N);
  if (use32) k_scatter<int><<<gE, T, 0, stream>>>(bufA, norm, s32, d32, E, bufB);
  else       k_scatter<long long><<<gE, T, 0, stream>>>(bufA, norm, ei, ei + E, E, bufB);
  k_relu4<<<gN4, T, 0, stream>>>((float4*)bufB, N * (LDF / 4));

  // MLP head: 30 -> 10 (relu, bias fused) -> 4 (bias fused)
  launch_gemm<32>(bufB, LDF, Wp4, 16, lb1p, bufA, 16, 16, N, 1, stream);
  launch_gemm<12>(bufA, 16, Wp5, 16, lb2p, out, 4, 4, N, 0, stream);
}